// SurfNet_15066745274908
// MI455X (gfx1250) — compile-verified
//
#include <hip/hip_runtime.h>

typedef __attribute__((ext_vector_type(2))) float v2f;
typedef __attribute__((ext_vector_type(8))) float v8f;

#define COL  512
#define TPAD 516   // padded LDS row stride (floats): 516 % 64 = 4 -> conflict-free column access

#if defined(__has_builtin)
#  if __has_builtin(__builtin_amdgcn_global_load_async_to_lds_b128) && \
      __has_builtin(__builtin_amdgcn_s_wait_asynccnt)
#    define USE_ASYNC 1
#  endif
#endif
#ifndef USE_ASYNC
#  define USE_ASYNC 0
#endif

#if USE_ASYNC
typedef int v4i __attribute__((vector_size(16)));
typedef __attribute__((address_space(1))) v4i gv4i;   // global int4*
typedef __attribute__((address_space(3))) v4i lv4i;   // LDS int4*
#endif

// One wave (32 threads) per block; each block handles 16 rows of length 512.
// Single HBM pass: async-stage tile to LDS, row min/max, WMMA moment GEMM, f64 fit.
__global__ __launch_bounds__(32) void surf_moments(const float* __restrict__ x,
                                                   float* __restrict__ muw,
                                                   float* __restrict__ sgw)
{
    __shared__ alignas(16) float tile[16 * TPAD];   // 33 KB staged logits
    __shared__ alignas(16) float btab[5 * TPAD];    // 10 KB Vandermonde basis t^n
    __shared__ float rminS[16], rmaxS[16];
    __shared__ float momA[16][8];
    __shared__ float momB[16][8];

    const int  lane    = threadIdx.x;               // 0..31
    const long rowBase = (long)blockIdx.x * 16;
    const float* gx    = x + rowBase * COL;
    const float sc     = 1.0f / 511.0f;             // column scaling t = k/511

    // ---- stage 16 rows (32 KB) into LDS ----
#if USE_ASYNC
    #pragma unroll 4
    for (int j = 0; j < 64; ++j) {                  // 32 lanes x 16 B = 512 B per issue
        const int r = j >> 2, q = j & 3;
        __builtin_amdgcn_global_load_async_to_lds_b128(
            (gv4i*)(gx + r * COL + q * 128 + lane * 4),
            (lv4i*)(tile + r * TPAD + q * 128 + lane * 4), 0, 0);
    }
#else
    #pragma unroll 4
    for (int j = 0; j < 64; ++j) {
        const int r = j >> 2, q = j & 3;
        const float4 v = *(const float4*)(gx + r * COL + q * 128 + lane * 4);
        *(float4*)(tile + r * TPAD + q * 128 + lane * 4) = v;
    }
#endif

    // ---- basis table fill (independent of tile -> overlaps the async DMA) ----
    #pragma unroll 4
    for (int i = 0; i < 16; ++i) {
        const int   k  = i * 32 + lane;
        const float t  = (float)k * sc;
        const float t2 = t * t, t3 = t2 * t, t4 = t2 * t2;
        btab[0 * TPAD + k] = 1.0f;
        btab[1 * TPAD + k] = t;
        btab[2 * TPAD + k] = t2;
        btab[3 * TPAD + k] = t3;
        btab[4 * TPAD + k] = t4;
    }

#if USE_ASYNC
    __builtin_amdgcn_s_wait_asynccnt(0);
#endif
    __syncthreads();

    // ---- per-row min/max: 2 lanes per row, one shfl_xor to combine ----
    {
        const int r2 = lane >> 1;                   // row 0..15
        const int h  = lane & 1;                    // which half of the row
        float vmin = 3.4e38f, vmax = -3.4e38f;
        #pragma unroll 8
        for (int j = 0; j < 64; ++j) {              // 64 x float4 = 256 floats
            const float4 v = *(const float4*)(tile + r2 * TPAD + h * 256 + j * 4);
            vmin = fminf(vmin, fminf(fminf(v.x, v.y), fminf(v.z, v.w)));
            vmax = fmaxf(vmax, fmaxf(fmaxf(v.x, v.y), fmaxf(v.z, v.w)));
        }
        vmin = fminf(vmin, __shfl_xor(vmin, 1, 32));
        vmax = fmaxf(vmax, __shfl_xor(vmax, 1, 32));
        if (h == 0) { rminS[r2] = vmin; rmaxS[r2] = vmax; }
    }
    __syncthreads();

    // ---- WMMA moment accumulation: D = A(16xK) * B(Kx16), K chunks of 4 ----
    // A (f32 16x4): lanes 0-15 hold row m=lane at K=kk+{0,1}; lanes 16-31 at K=kk+{2,3}.
    // B (f32 4x16): lane n holds basis_n at the same K pair (from btab; n>=5 -> row 4,
    //               those D columns are never read).
    const int   m    = lane & 15;
    const int   koff = (lane >> 4) * 2;
    const float LOG2E = 1.4426950408889634f;
    const float M    = rmaxS[m];
    const float emin = __builtin_amdgcn_exp2f((rminS[m] - M) * LOG2E);
    const float k1   = 1.0f / (1.0f - emin);
    const float k0   = 0.001f - emin * k1;          // y = e^(x-M)*k1 + k0
    const float eb   = -M * LOG2E;
    const float* trow = tile + m * TPAD + koff;
    const float* brow = btab + (m < 5 ? m : 4) * TPAD + koff;

    v8f cA = {0.f, 0.f, 0.f, 0.f, 0.f, 0.f, 0.f, 0.f};
    v8f cB = {0.f, 0.f, 0.f, 0.f, 0.f, 0.f, 0.f, 0.f};

    #pragma unroll 4
    for (int kk = 0; kk < COL; kk += 4) {
        const v2f xv = *(const v2f*)(trow + kk);
        const v2f B  = *(const v2f*)(brow + kk);
        float e0 = __builtin_amdgcn_exp2f(fmaf(xv.x, LOG2E, eb));
        float e1 = __builtin_amdgcn_exp2f(fmaf(xv.y, LOG2E, eb));
        float y0 = fmaf(e0, k1, k0);
        float y1 = fmaf(e1, k1, k0);
        float y20 = y0 * y0, y21 = y1 * y1;
        v2f A1 = { y20, y21 };                               // y^2
        // y^2 * log2(y): the ln2 factor is folded into the f64 moment un-scaling
        v2f A2 = { y20 * __builtin_amdgcn_logf(y0), y21 * __builtin_amdgcn_logf(y1) };
        cA = __builtin_amdgcn_wmma_f32_16x16x4_f32(false, A1, false, B, (short)0, cA, false, false);
        cB = __builtin_amdgcn_wmma_f32_16x16x4_f32(false, A2, false, B, (short)0, cB, false, false);
    }

    // ---- scatter D (moments) to LDS: D[m][n], lanes 0-15 -> rows 0..7, 16-31 -> rows 8..15
    const int hi = lane >> 4;
    if (m < 5) {
        #pragma unroll
        for (int r = 0; r < 8; ++r) momA[hi * 8 + r][m] = cA[r];
    }
    if (m < 3) {
        #pragma unroll
        for (int r = 0; r < 8; ++r) momB[hi * 8 + r][m] = cB[r];
    }
    __syncthreads();

    // ---- per-row closed-form Gaussian fit in f64 (matches reference float64 math) ----
    if (lane < 16) {
        const double c1 = 511.0, c2 = c1 * c1, c3 = c2 * c1, c4 = c2 * c2;
        const double LN2 = 0.6931471805599453;
        double s_y2    = (double)momA[lane][0];
        double s_xy2   = (double)momA[lane][1] * c1;   // un-scale t = k/511 moments
        double s_x2y2  = (double)momA[lane][2] * c2;
        double s_x3y2  = (double)momA[lane][3] * c3;
        double s_x4y2  = (double)momA[lane][4] * c4;
        double s_y2l   = (double)momB[lane][0] * LN2;  // log2 -> ln
        double s_xy2l  = (double)momB[lane][1] * (c1 * LN2);
        double s_x2y2l = (double)momB[lane][2] * (c2 * LN2);

        double b_num = s_x2y2 * s_x2y2 * s_xy2l - s_y2 * s_x4y2 * s_xy2l
                     + s_xy2 * s_x4y2 * s_y2l  + s_y2 * s_x3y2 * s_x2y2l
                     - s_x2y2 * s_x3y2 * s_y2l - s_xy2 * s_x2y2 * s_x2y2l;
        double c_num = s_x2y2l * s_xy2 * s_xy2 - s_xy2l * s_xy2 * s_x2y2
                     - s_x3y2 * s_y2l * s_xy2  + s_y2l * s_x2y2 * s_x2y2
                     - s_y2 * s_x2y2l * s_x2y2 + s_y2 * s_x3y2 * s_xy2l;
        if (fabs(c_num) < 1e-15) {
            double sg = (c_num > 0.0) ? 1.0 : ((c_num < 0.0) ? -1.0 : 0.0);
            c_num = sg * 1e-15;
        }
        double mu = -b_num / (2.0 * c_num);
        double c_din = s_x4y2 * s_xy2 * s_xy2 - 2.0 * s_xy2 * s_x2y2 * s_x3y2
                     + s_x2y2 * s_x2y2 * s_x2y2 - s_y2 * s_x4y2 * s_x2y2
                     + s_y2 * s_x3y2 * s_x3y2;
        double sigma = -0.5 * c_din / c_num;
        if (sigma < 1.0) sigma = 1.0;

        muw[rowBase + lane] = (float)mu;
        sgw[rowBase + lane] = (float)sigma;
    }
}

// Thomas algorithm on the tridiagonal CRF system, one thread per batch.
// In-place over the mu/sigma workspace (cp overwrites sigma, dp overwrites mu).
__global__ void surf_solve(const float* __restrict__ wc,
                           float* __restrict__ muw,
                           float* __restrict__ sgw,
                           float* __restrict__ out, int bn)
{
    const int b = blockIdx.x * blockDim.x + threadIdx.x;
    if (b >= bn) return;
    const int n   = 1024;
    const float w   = wc[0];
    const float off = -2.0f * w;
    float* mu = muw + (long)b * n;
    float* sg = sgw + (long)b * n;

    float sr   = 1.0f / sg[0];
    float diag = 2.0f * w + sr;
    float cp   = off / diag;
    float dp   = (mu[0] * sr) / diag;
    sg[0] = cp; mu[0] = dp;
    for (int i = 1; i < n; ++i) {
        sr = 1.0f / sg[i];
        float main = ((i == n - 1) ? 2.0f * w : 4.0f * w) + sr;
        float d    = mu[i] * sr;
        float den  = main - off * cp;
        cp = off / den;
        dp = (d - off * dp) / den;
        sg[i] = cp; mu[i] = dp;
    }
    float xv = dp;
    out[(long)b * n + n - 1] = xv;
    for (int i = n - 2; i >= 0; --i) {
        xv = mu[i] - sg[i] * xv;
        out[(long)b * n + i] = xv;
    }
}

extern "C" void kernel_launch(void* const* d_in, const int* in_sizes, int n_in,
                              void* d_out, int out_size, void* d_ws, size_t ws_size,
                              hipStream_t stream) {
    const float* x  = (const float*)d_in[0];
    const float* wc = (const float*)d_in[1];
    float* out = (float*)d_out;

    const int rows = in_sizes[0] / COL;    // 32768
    const int bn   = rows / 1024;          // 32

    float* muw = (float*)d_ws;             // rows floats
    float* sgw = muw + rows;               // rows floats

    surf_moments<<<dim3(rows / 16), dim3(32), 0, stream>>>(x, muw, sgw);
    surf_solve<<<dim3(1), dim3(bn), 0, stream>>>(wc, muw, sgw, out, bn);
}